// AtomMessagePassing_57921928954076
// MI455X (gfx1250) — compile-verified
//
#include <hip/hip_runtime.h>

#define N_NODES 50000
#define N_EDGES 800000
#define DV 133
#define DE 14
#define DH 300

#define MP 50048   // N_NODES padded to 64 (782 * 64)
#define WP 304     // weight-row count padded to 16
#define KVP 136    // DV padded to multiple of 8

typedef __attribute__((ext_vector_type(2))) float v2f;
typedef __attribute__((ext_vector_type(8))) float v8f;

// ---------------------------------------------------------------------------
// Zero-padded pack: dst[Mp x Kp] <- src[Msrc x Ksrc] (row stride srcLd, col
// offset colOff), zeros elsewhere. Makes all GEMM operand loads guard-free.
// ---------------------------------------------------------------------------
__global__ __launch_bounds__(256) void pack_pad(const float* __restrict__ src,
                                                int srcLd, int colOff,
                                                int Msrc, int Ksrc,
                                                float* __restrict__ dst,
                                                int Mp, int Kp) {
  int gid = blockIdx.x * blockDim.x + threadIdx.x;
  if (gid >= Mp * Kp) return;
  int r = gid / Kp;
  int c = gid - r * Kp;
  float v = (r < Msrc && c < Ksrc) ? src[(size_t)r * srcLd + colOff + c] : 0.0f;
  dst[gid] = v;
}

// ---------------------------------------------------------------------------
// Edge aggregation of bond features: Eagg[dst] += E[e], deg[dst] += 1
// ---------------------------------------------------------------------------
__global__ __launch_bounds__(256) void edge_agg_e(const float* __restrict__ E,
                                                  const long long* __restrict__ ei,
                                                  float* __restrict__ Eagg,
                                                  float* __restrict__ deg) {
  int gid = blockIdx.x * blockDim.x + threadIdx.x;
  if (gid >= N_EDGES * 16) return;
  int e = gid >> 4;
  int f = gid & 15;
  int dst = (int)ei[N_EDGES + e];
  if (f < DE) {
    atomicAdd(&Eagg[dst * DE + f], E[e * DE + f]);
  } else if (f == DE) {
    atomicAdd(&deg[dst], 1.0f);
  }
}

// ---------------------------------------------------------------------------
// S[dst] += H[src] per edge. One wave per edge; lanes stride the row in
// float4 units (rows are 1200 B, 16 B aligned) -> global_load_b128 gathers.
// ---------------------------------------------------------------------------
__global__ __launch_bounds__(256) void scatter_rows(const float* __restrict__ H,
                                                    const long long* __restrict__ ei,
                                                    float* __restrict__ S) {
  int w = blockIdx.x * (blockDim.x >> 5) + (threadIdx.x >> 5);
  if (w >= N_EDGES) return;
  int lane = threadIdx.x & 31;
  int src = (int)ei[w];
  int dst = (int)ei[N_EDGES + w];
  const float4* h = (const float4*)(H + (size_t)src * DH);
  float* s = S + (size_t)dst * DH;
  for (int j = lane; j < DH / 4; j += 32) {
    float4 v = h[j];
    atomicAdd(&s[4 * j + 0], v.x);
    atomicAdd(&s[4 * j + 1], v.y);
    atomicAdd(&s[4 * j + 2], v.z);
    atomicAdd(&s[4 * j + 3], v.w);
  }
}

// ---------------------------------------------------------------------------
// B0[n][j] = H0[n][j] + deg[n]*b_h[j] + sum_k Eagg[n][k] * W_h[j][300+k]
// In-place over H0. One block per node.
// ---------------------------------------------------------------------------
__global__ __launch_bounds__(320) void b0_fuse(const float* __restrict__ Eagg,
                                               const float* __restrict__ deg,
                                               const float* __restrict__ W_h,
                                               const float* __restrict__ b_h,
                                               float* __restrict__ H0B0) {
  __shared__ float er[DE + 1];
  int n = blockIdx.x;
  int t = threadIdx.x;
  if (t < DE) er[t] = Eagg[n * DE + t];
  if (t == DE) er[DE] = deg[n];
  __syncthreads();
  for (int j = t; j < DH; j += blockDim.x) {
    float acc = er[DE] * b_h[j] + H0B0[(size_t)n * DH + j];
#pragma unroll
    for (int k = 0; k < DE; ++k) acc += er[k] * W_h[j * (DE + DH) + DH + k];
    H0B0[(size_t)n * DH + j] = acc;
  }
}

// ---------------------------------------------------------------------------
// Software-pipelined guard-free WMMA fp32 panel. Fragments for step i+1 are
// loaded before step i's WMMAs consume their (iteration-old) fragments, so
// waits are on loads issued a full iteration earlier.
// ---------------------------------------------------------------------------
#define WMMA_F32(a, b, cc) \
  __builtin_amdgcn_wmma_f32_16x16x4_f32(false, (a), false, (b), (short)0, (cc), false, false)

__device__ __forceinline__ void mma_panel(const float* __restrict__ A, int lda,
                                          const float* __restrict__ W, int ldw, int Kp,
                                          int mBase, int nColP, int l, int hi, v8f* c) {
  const float* wr = W + (size_t)nColP * ldw + 2 * hi;
  const float* a0 = A + (size_t)(mBase + l) * lda + 2 * hi;
  const size_t ts = (size_t)16 * lda;

  v2f b = *(const v2f*)(wr);
  v2f x0 = *(const v2f*)(a0);
  v2f x1 = *(const v2f*)(a0 + ts);
  v2f x2 = *(const v2f*)(a0 + 2 * ts);
  v2f x3 = *(const v2f*)(a0 + 3 * ts);

#pragma unroll 2
  for (int k0 = 4; k0 < Kp; k0 += 4) {
    v2f bn = *(const v2f*)(wr + k0);
    v2f y0 = *(const v2f*)(a0 + k0);
    v2f y1 = *(const v2f*)(a0 + ts + k0);
    v2f y2 = *(const v2f*)(a0 + 2 * ts + k0);
    v2f y3 = *(const v2f*)(a0 + 3 * ts + k0);
    c[0] = WMMA_F32(x0, b, c[0]);
    c[1] = WMMA_F32(x1, b, c[1]);
    c[2] = WMMA_F32(x2, b, c[2]);
    c[3] = WMMA_F32(x3, b, c[3]);
    b = bn;
    x0 = y0; x1 = y1; x2 = y2; x3 = y3;
  }
  c[0] = WMMA_F32(x0, b, c[0]);
  c[1] = WMMA_F32(x1, b, c[1]);
  c[2] = WMMA_F32(x2, b, c[2]);
  c[3] = WMMA_F32(x3, b, c[3]);
}

// D = relu( A1 @ W1^T [+ A2 @ W2^T] [+ bias] [+ addMat] ); guards only in epilogue.
__global__ __launch_bounds__(256) void gemm_wmma(
    const float* __restrict__ A1, int lda1, const float* __restrict__ W1, int ldw1, int K1p,
    const float* __restrict__ A2, int lda2, const float* __restrict__ W2, int ldw2, int K2p,
    const float* __restrict__ bias, const float* __restrict__ addMat,
    float* __restrict__ D, int M, int Nout, int mt64, int doRelu) {
  const int wid = blockIdx.x * (blockDim.x >> 5) + (threadIdx.x >> 5);
  const int ntiles = (Nout + 15) >> 4;
  const int nt = wid / mt64;
  const int mt = wid - nt * mt64;
  if (nt >= ntiles) return;

  const int lane = threadIdx.x & 31;
  const int l = lane & 15;
  const int hi = lane >> 4;
  const int mBase = mt * 64;
  const int nCol = nt * 16 + l;
  const bool nOK = (nCol < Nout);

  v8f c[4] = {};
  mma_panel(A1, lda1, W1, ldw1, K1p, mBase, nCol, l, hi, c);
  if (A2) mma_panel(A2, lda2, W2, ldw2, K2p, mBase, nCol, l, hi, c);

  float badd = (bias && nOK) ? bias[nCol] : 0.0f;
#pragma unroll
  for (int t = 0; t < 4; ++t) {
#pragma unroll
    for (int r = 0; r < 8; ++r) {
      const int row = mBase + 16 * t + 8 * hi + r;
      if (row < M && nOK) {
        float v = c[t][r] + badd;
        if (addMat) v += addMat[(size_t)row * Nout + nCol];
        if (doRelu) v = fmaxf(v, 0.0f);
        D[(size_t)row * Nout + nCol] = v;
      }
    }
  }
}

// ---------------------------------------------------------------------------
extern "C" void kernel_launch(void* const* d_in, const int* in_sizes, int n_in,
                              void* d_out, int out_size, void* d_ws, size_t ws_size,
                              hipStream_t stream) {
  const float* V = (const float*)d_in[0];
  const float* E = (const float*)d_in[1];
  const long long* EI = (const long long*)d_in[2];
  const float* W_i = (const float*)d_in[3];
  const float* b_i = (const float*)d_in[4];
  const float* W_h = (const float*)d_in[5];
  const float* b_h = (const float*)d_in[6];
  const float* W_o = (const float*)d_in[7];
  const float* b_o = (const float*)d_in[8];
  float* out = (float*)d_out;

  float* ws = (float*)d_ws;
  size_t o = 0;
  float* S = ws + o;    o += (size_t)MP * DH;          // padded rows: A-input
  float* B0 = ws + o;   o += (size_t)N_NODES * DH;     // H0 -> fused B0
  float* H = ws + o;    o += (size_t)N_NODES * DH;
  float* Vp = ws + o;   o += (size_t)MP * KVP;         // padded V copy
  float* Wip = ws + o;  o += (size_t)WP * KVP;
  float* Whhp = ws + o; o += (size_t)WP * DH;
  float* Wovp = ws + o; o += (size_t)WP * KVP;
  float* Wohp = ws + o; o += (size_t)WP * DH;
  float* Eagg = ws + o; o += (size_t)N_NODES * DE;
  float* deg = ws + o;  o += (size_t)N_NODES;

  const int mt64 = MP / 64;                        // 782
  const int ntiles = (DH + 15) / 16;               // 19
  const int gemmBlocks = (mt64 * ntiles + 7) / 8;  // 8 waves / block
  const int scatBlocks = (N_EDGES + 7) / 8;

  // 0) pack zero-padded operands (guard-free GEMM loads)
  {
    int n1 = MP * KVP;
    pack_pad<<<(n1 + 255) / 256, 256, 0, stream>>>(V, DV, 0, N_NODES, DV, Vp, MP, KVP);
    int n2 = WP * KVP;
    pack_pad<<<(n2 + 255) / 256, 256, 0, stream>>>(W_i, DV, 0, DH, DV, Wip, WP, KVP);
    int n3 = WP * DH;
    pack_pad<<<(n3 + 255) / 256, 256, 0, stream>>>(W_h, DE + DH, 0, DH, DH, Whhp, WP, DH);
    pack_pad<<<(n2 + 255) / 256, 256, 0, stream>>>(W_o, DV + DH, 0, DH, DV, Wovp, WP, KVP);
    pack_pad<<<(n3 + 255) / 256, 256, 0, stream>>>(W_o, DV + DH, DV, DH, DH, Wohp, WP, DH);
  }

  // 1) aggregate bond features + degree
  hipMemsetAsync(Eagg, 0, (size_t)N_NODES * (DE + 1) * sizeof(float), stream);
  edge_agg_e<<<(N_EDGES * 16 + 255) / 256, 256, 0, stream>>>(E, EI, Eagg, deg);

  // 2) H0 = relu(Vp @ Wip^T + b_i) -> B0 buffer
  gemm_wmma<<<gemmBlocks, 256, 0, stream>>>(Vp, KVP, Wip, KVP, KVP,
                                            nullptr, 0, nullptr, 0, 0,
                                            b_i, nullptr, B0, N_NODES, DH, mt64, 1);

  // 3) layer-1 gather from H0 (before in-place B0 fusion)
  hipMemsetAsync(S, 0, (size_t)MP * DH * sizeof(float), stream);
  scatter_rows<<<scatBlocks, 256, 0, stream>>>(B0, EI, S);

  // 4) B0 = H0 + Eagg @ W_he^T + deg * b_h (in-place)
  b0_fuse<<<N_NODES, 320, 0, stream>>>(Eagg, deg, W_h, b_h, B0);

  // 5) H = relu(S @ Whh^T + B0)
  gemm_wmma<<<gemmBlocks, 256, 0, stream>>>(S, DH, Whhp, DH, DH,
                                            nullptr, 0, nullptr, 0, 0,
                                            nullptr, B0, H, N_NODES, DH, mt64, 1);

  // 6) second message pass
  hipMemsetAsync(S, 0, (size_t)MP * DH * sizeof(float), stream);
  scatter_rows<<<scatBlocks, 256, 0, stream>>>(H, EI, S);
  gemm_wmma<<<gemmBlocks, 256, 0, stream>>>(S, DH, Whhp, DH, DH,
                                            nullptr, 0, nullptr, 0, 0,
                                            nullptr, B0, H, N_NODES, DH, mt64, 1);

  // 7) final aggregation M_v
  hipMemsetAsync(S, 0, (size_t)MP * DH * sizeof(float), stream);
  scatter_rows<<<scatBlocks, 256, 0, stream>>>(H, EI, S);

  // 8) out = relu(Vp @ Wovp^T + M_v @ Wohp^T + b_o)
  gemm_wmma<<<gemmBlocks, 256, 0, stream>>>(Vp, KVP, Wovp, KVP, KVP,
                                            S, DH, Wohp, DH, DH,
                                            b_o, nullptr, out, N_NODES, DH, mt64, 1);
}